// LinkPredictor_16217796510234
// MI455X (gfx1250) — compile-verified
//
#include <hip/hip_runtime.h>
#include <hip/hip_bf16.h>
#include <math.h>

#define D_FEAT   128
#define N_NODES  100000
#define N_EDGES  1000000

typedef __attribute__((ext_vector_type(2))) float v2f;
typedef __attribute__((ext_vector_type(8))) float v8f;

// ---------------------------------------------------------------------------
// Pass 1: per-node projections via V_WMMA_F32_16X16X4_F32.
// One wave = one 16-node tile. C(16x16) accumulates A(16xK) x B(Kx16) over
// K=128 in 32 steps of K=4. B columns 0/1 hold W_src/W_dst, columns 2..15 are
// zero. Result: s[node] in C column 0, t[node] in C column 1.
//
// A 16x4 f32 layout (per ISA): lanes 0-15 hold M=lane, VGPR0=K0,VGPR1=K1;
// lanes 16-31 hold M=lane-16, VGPR0=K2,VGPR1=K3  -> one b64 load per lane.
// C 16x16 f32 layout: VGPR v, lanes 0-15: D[M=v][N=lane];
// lanes 16-31: D[M=v+8][N=lane-16].
// ---------------------------------------------------------------------------
__global__ __launch_bounds__(128) void proj_wmma_kernel(
    const float* __restrict__ z, const float* __restrict__ W,
    float* __restrict__ st) {
  const int wave = (int)((blockIdx.x * blockDim.x + threadIdx.x) >> 5);
  const int lane = (int)(threadIdx.x & 31);
  const int ntiles = N_NODES / 16;  // 6250, exact
  if (wave >= ntiles) return;       // wave-uniform: EXEC stays all-ones inside

  const int row  = lane & 15;       // A: M index; B/C: N index
  const int half = lane >> 4;       // 0 -> K{0,1}, 1 -> K{2,3}

  const float* __restrict__ arow  = z + (size_t)(wave * 16 + row) * D_FEAT;
  const bool   wvalid             = (row < 2);
  const float* __restrict__ wbase = wvalid ? (W + row * D_FEAT) : W;

  v8f c = {};
#pragma unroll
  for (int k0 = 0; k0 < D_FEAT; k0 += 4) {
    const int kk = k0 + 2 * half;
    v2f a  = *(const v2f*)(arow + kk);    // A[M=row][kk], A[M=row][kk+1]
    v2f wv = *(const v2f*)(wbase + kk);   // W column data (hot in L0)
    v2f b;
    b.x = wvalid ? wv.x : 0.0f;           // v_cndmask, no EXEC change
    b.y = wvalid ? wv.y : 0.0f;
    // 8 args: (neg_a, A, neg_b, B, c_mod, C, reuse_a, reuse_b)
    c = __builtin_amdgcn_wmma_f32_16x16x4_f32(false, a, false, b,
                                              (short)0, c, false, false);
  }

  // s -> lanes N=0 (0,16), t -> lanes N=1 (1,17); each holds 8 M values.
  if (row < 2) {
    const int mbase = wave * 16 + half * 8;
#pragma unroll
    for (int v = 0; v < 8; ++v) {
      st[2 * (mbase + v) + row] = c[v];
    }
  }
}

// ---------------------------------------------------------------------------
// Pass 2: out[e] = sigmoid(s[src] + t[dst] + b). 800KB table gathers hit L2.
// ---------------------------------------------------------------------------
__global__ __launch_bounds__(256) void edge_sigmoid_kernel(
    const int* __restrict__ ei, const float* __restrict__ st,
    const float* __restrict__ bias, float* __restrict__ out) {
  const int e = (int)(blockIdx.x * blockDim.x + threadIdx.x);
  if (e >= N_EDGES) return;
  const int s = ei[e];             // src row of edge_index
  const int d = ei[N_EDGES + e];   // dst row of edge_index
  const float logit = st[2 * s] + st[2 * d + 1] + bias[0];
  out[e] = 1.0f / (1.0f + expf(-logit));
}

extern "C" void kernel_launch(void* const* d_in, const int* in_sizes, int n_in,
                              void* d_out, int out_size, void* d_ws,
                              size_t ws_size, hipStream_t stream) {
  (void)in_sizes; (void)n_in; (void)out_size; (void)ws_size;
  const float* z  = (const float*)d_in[0];
  const int*   ei = (const int*)d_in[1];   // integer inputs -> const int*
  const float* W  = (const float*)d_in[2];
  const float* b  = (const float*)d_in[3];
  float* st  = (float*)d_ws;               // 100000 x 2 floats = 800 KB
  float* out = (float*)d_out;

  const int ntiles  = N_NODES / 16;                 // 6250 waves
  const int blocks1 = (ntiles + 3) / 4;             // 4 waves (128 thr) / block
  proj_wmma_kernel<<<blocks1, 128, 0, stream>>>(z, W, st);

  const int blocks2 = (N_EDGES + 255) / 256;
  edge_sigmoid_kernel<<<blocks2, 256, 0, stream>>>(ei, st, b, out);
}